// DeepDEQAutoencoder_4578435137818
// MI455X (gfx1250) — compile-verified
//
#include <hip/hip_runtime.h>

// ---------------------------------------------------------------------------
// DEQ autoencoder with Anderson acceleration, fp32, CDNA5 (gfx1250).
// GEMMs use V_WMMA_F32_16X16X4_F32 (native fp32 matrix op, exact precision).
// LDS tiles are k-pair-interleaved (float2) so each WMMA fragment is a single
// ds_load_b64 into an even-aligned VGPR pair (no v_mov repacking).
// ---------------------------------------------------------------------------

#define BATCH 4096
#define D0    1024
#define D1    512
#define D2    512
#define D3    512
#define DTOT  1536            // D1+D2+D3
#define M_AA  5
#define LAM   1.0e-4f
#define THRES 50
#define OUTW  3584            // 1024+512+512+1536

typedef float v2f __attribute__((ext_vector_type(2)));
typedef float v8f __attribute__((ext_vector_type(8)));

// ------------------------------ WMMA GEMM ----------------------------------
// C[M,N] = op(A[M,K] * B) (+bias) (relu?)   A row-major stride lda.
// BMODE 0: B is N x K row-major (TN gemm, C = A * B^T)   -- encoder
// BMODE 1: B is K x N row-major (NN gemm, C = A * B)     -- decoder
// Block tile 128x128, K-chunk 16. 8 waves: 2 along M (64 rows) x 4 along N
// (32 cols). Per wave: 4x2 = 8 accumulator tiles of 16x16.
#define MBLK 128
#define NBLK 128
#define KBLK 16
#define LPAD 16   // float2 row pad: (128+16)*2 banks mod 64 == 32 -> the two
                  // half-wave fragment reads land in disjoint bank halves

template <int BMODE, bool RELU, bool BIAS>
__global__ __launch_bounds__(256) void gemm_wmma(
    const float* __restrict__ A, int lda,
    const float* __restrict__ Bw, int ldb,
    float* __restrict__ C, int ldc,
    const float* __restrict__ bias, int K)
{
  __shared__ v2f As2[KBLK / 2][MBLK + LPAD];
  __shared__ v2f Bs2[KBLK / 2][NBLK + LPAD];

  const int tid  = threadIdx.x;
  const int lane = tid & 31;
  const int wave = tid >> 5;      // 0..7
  const int wm   = wave & 1;      // 2 waves along M (64 rows each)
  const int wn   = wave >> 1;     // 4 waves along N (32 cols each)
  const int m0   = blockIdx.y * MBLK;
  const int n0   = blockIdx.x * NBLK;
  const int half = lane >> 4;     // WMMA half-wave (K pair select)
  const int l16  = lane & 15;

  // staging thread mapping (A, and B in TN mode): row r = tid&127,
  // k-quarter q = (tid>>7) + 2h, h in {0,1}; each load is a float4 (4 k's).
  const int srow = tid & 127;
  const int q0   = tid >> 7;
  // staging mapping for B in NN mode: k row kb = tid>>5 (+8h), 4 cols each.
  const int kb   = tid >> 5;
  const int n4   = (tid & 31) * 4;

  auto loadA = [&](int k0, float4* dst) {
    #pragma unroll
    for (int h = 0; h < 2; ++h) {
      const int kk = (q0 + 2 * h) * 4;
      dst[h] = *(const float4*)(A + (size_t)(m0 + srow) * lda + k0 + kk);
    }
  };
  auto loadB = [&](int k0, float4* dst) {
    if (BMODE == 0) {
      #pragma unroll
      for (int h = 0; h < 2; ++h) {
        const int kk = (q0 + 2 * h) * 4;
        dst[h] = *(const float4*)(Bw + (size_t)(n0 + srow) * ldb + k0 + kk);
      }
    } else {
      #pragma unroll
      for (int h = 0; h < 2; ++h) {
        const int k = kb + 8 * h;
        dst[h] = *(const float4*)(Bw + (size_t)(k0 + k) * ldb + n0 + n4);
      }
    }
  };
  auto storeA = [&](const float4* src) {
    #pragma unroll
    for (int h = 0; h < 2; ++h) {
      const int q = q0 + 2 * h;          // k = 4q .. 4q+3 -> pair rows 2q,2q+1
      v2f lo = {src[h].x, src[h].y};
      v2f hi = {src[h].z, src[h].w};
      As2[2 * q + 0][srow] = lo;
      As2[2 * q + 1][srow] = hi;
    }
  };
  auto storeB = [&](const float4* src) {
    if (BMODE == 0) {
      #pragma unroll
      for (int h = 0; h < 2; ++h) {
        const int q = q0 + 2 * h;
        v2f lo = {src[h].x, src[h].y};
        v2f hi = {src[h].z, src[h].w};
        Bs2[2 * q + 0][srow] = lo;
        Bs2[2 * q + 1][srow] = hi;
      }
    } else {
      #pragma unroll
      for (int h = 0; h < 2; ++h) {
        const int k = kb + 8 * h;
        const float4 v = src[h];
        Bs2[k >> 1][n4 + 0][k & 1] = v.x;
        Bs2[k >> 1][n4 + 1][k & 1] = v.y;
        Bs2[k >> 1][n4 + 2][k & 1] = v.z;
        Bs2[k >> 1][n4 + 3][k & 1] = v.w;
      }
    }
  };

  v8f acc[4][2];
  #pragma unroll
  for (int i = 0; i < 4; ++i)
    #pragma unroll
    for (int j = 0; j < 2; ++j) acc[i][j] = (v8f){};

  float4 apre[2], bpre[2];
  loadA(0, apre);
  loadB(0, bpre);

  for (int k0 = 0; k0 < K; k0 += KBLK) {
    storeA(apre);
    storeB(bpre);
    __syncthreads();
    if (k0 + KBLK < K) {               // prefetch next chunk during compute
      loadA(k0 + KBLK, apre);
      loadB(k0 + KBLK, bpre);
    }
    // 4 WMMA k-steps of depth 4; fragment = one float2 at pair-row kk2+half
    #pragma unroll
    for (int kk2 = 0; kk2 < 8; kk2 += 2) {
      const int krow = kk2 + half;
      v2f afr[4], bfr[2];
      #pragma unroll
      for (int tm = 0; tm < 4; ++tm)
        afr[tm] = As2[krow][wm * 64 + tm * 16 + l16];
      #pragma unroll
      for (int tn = 0; tn < 2; ++tn)
        bfr[tn] = Bs2[krow][wn * 32 + tn * 16 + l16];
      #pragma unroll
      for (int tm = 0; tm < 4; ++tm)
        #pragma unroll
        for (int tn = 0; tn < 2; ++tn)
          acc[tm][tn] = __builtin_amdgcn_wmma_f32_16x16x4_f32(
              false, afr[tm], false, bfr[tn], (short)0, acc[tm][tn],
              false, false);
    }
    __syncthreads();
  }

  // --- epilogue: C/D layout row = r + 8*half, col = l16
  #pragma unroll
  for (int tn = 0; tn < 2; ++tn) {
    const int col = n0 + wn * 32 + tn * 16 + l16;
    const float bv = BIAS ? bias[col] : 0.0f;
    #pragma unroll
    for (int tm = 0; tm < 4; ++tm) {
      #pragma unroll
      for (int r = 0; r < 8; ++r) {
        const int row = m0 + wm * 64 + tm * 16 + r + half * 8;
        float v = acc[tm][tn][r] + bv;
        if (RELU) v = fmaxf(v, 0.0f);
        C[(size_t)row * ldc + col] = v;
      }
    }
  }
}

// --------------------------- Anderson step ---------------------------------
// One wave (32 lanes) per batch row. Computes Gram of G=F-X over the n live
// history slots, solves H w = 1 (H SPD + lam*I), alpha = w/sum(w) (== bordered
// solve with sum(alpha)=1), writes x_k = sum alpha_i F_i into X[slot], and
// stamps the constant n1 block into F[slot][0:512].
__global__ __launch_bounds__(256) void anderson_step(
    float* __restrict__ F, float* __restrict__ X,
    const float* __restrict__ N1, int n, int slot)
{
  const int lane = threadIdx.x & 31;
  const int wave = threadIdx.x >> 5;
  const int b = blockIdx.x * 8 + wave;
  const size_t base = (size_t)b * (M_AA * DTOT);

  float S[15];
  #pragma unroll
  for (int i = 0; i < 15; ++i) S[i] = 0.0f;

  for (int d = lane; d < DTOT; d += 32) {
    float g[M_AA];
    #pragma unroll
    for (int i = 0; i < M_AA; ++i)
      g[i] = (i < n) ? (F[base + i * DTOT + d] - X[base + i * DTOT + d]) : 0.0f;
    int p = 0;
    #pragma unroll
    for (int i = 0; i < M_AA; ++i)
      #pragma unroll
      for (int j = i; j < M_AA; ++j) { S[p] += g[i] * g[j]; ++p; }
  }
  // wave32 all-reduce of the 15 Gram entries
  #pragma unroll
  for (int p = 0; p < 15; ++p) {
    float v = S[p];
    #pragma unroll
    for (int off = 16; off > 0; off >>= 1) v += __shfl_xor(v, off, 32);
    S[p] = v;
  }

  // H = GG^T + lam*I; solve H w = 1 (unpivoted GE, SPD), alpha = w / sum(w)
  float Hm[M_AA][M_AA], w[M_AA], al[M_AA];
  {
    int p = 0;
    #pragma unroll
    for (int i = 0; i < M_AA; ++i) {
      #pragma unroll
      for (int j = i; j < M_AA; ++j) { Hm[i][j] = S[p]; Hm[j][i] = S[p]; ++p; }
      Hm[i][i] += LAM;
    }
  }
  #pragma unroll
  for (int i = 0; i < M_AA; ++i) w[i] = (i < n) ? 1.0f : 0.0f;
  #pragma unroll
  for (int p = 0; p < M_AA; ++p) {
    if (p < n) {
      const float inv = 1.0f / Hm[p][p];
      #pragma unroll
      for (int r = p + 1; r < M_AA; ++r) {
        if (r < n) {
          const float fm = Hm[r][p] * inv;
          #pragma unroll
          for (int c = p; c < M_AA; ++c) Hm[r][c] -= fm * Hm[p][c];
          w[r] -= fm * w[p];
        }
      }
    }
  }
  #pragma unroll
  for (int p = M_AA - 1; p >= 0; --p) {
    float s = w[p];
    #pragma unroll
    for (int c = p + 1; c < M_AA; ++c) if (c < n) s -= Hm[p][c] * al[c];
    al[p] = (p < n) ? (s / Hm[p][p]) : 0.0f;
  }
  float asum = 0.0f;
  #pragma unroll
  for (int i = 0; i < M_AA; ++i) if (i < n) asum += al[i];
  const float rs = 1.0f / asum;
  #pragma unroll
  for (int i = 0; i < M_AA; ++i) al[i] *= rs;

  // x_k = sum alpha_i * F_i  (beta == 1.0)  -> X[slot]
  for (int d = lane; d < DTOT; d += 32) {
    float acc = 0.0f;
    #pragma unroll
    for (int i = 0; i < M_AA; ++i)
      if (i < n) acc += al[i] * F[base + i * DTOT + d];
    X[base + (size_t)slot * DTOT + d] = acc;
  }
  // F[slot][0:D1] = n1 (constant across iterations); GEMMs fill the rest
  for (int d = lane; d < D1; d += 32)
    F[base + (size_t)slot * DTOT + d] = N1[(size_t)b * D1 + d];
}

// --------------------------- init / copy -----------------------------------
__global__ void init_hist(float* __restrict__ X, float* __restrict__ F,
                          const float* __restrict__ N1)
{
  const size_t idx = (size_t)blockIdx.x * blockDim.x + threadIdx.x;
  if (idx >= (size_t)BATCH * DTOT) return;
  const int b = (int)(idx / DTOT);
  const int d = (int)(idx % DTOT);
  const float n1v = (d < D1) ? N1[(size_t)b * D1 + d] : 0.0f;
  const size_t base = (size_t)b * (M_AA * DTOT);
  X[base + d]        = 0.0f;   // X[:,0] = x0 = 0
  F[base + d]        = n1v;    // F[:,0] = f(0) = [n1, 0, 0]
  X[base + DTOT + d] = n1v;    // X[:,1] = f(0)
  F[base + DTOT + d] = (d < D1) ? n1v : 0.0f;  // F[:,1] = [n1, gemm, 0]
}

__global__ void copy_zstar(const float* __restrict__ F, float* __restrict__ out)
{
  const size_t idx = (size_t)blockIdx.x * blockDim.x + threadIdx.x;
  if (idx >= (size_t)BATCH * DTOT) return;
  const int b = (int)(idx / DTOT);
  const int d = (int)(idx % DTOT);
  out[(size_t)b * OUTW + 2048 + d] =
      F[(size_t)b * (M_AA * DTOT) + (size_t)4 * DTOT + d];
}

// --------------------------- launch ----------------------------------------
extern "C" void kernel_launch(void* const* d_in, const int* in_sizes, int n_in,
                              void* d_out, int out_size, void* d_ws, size_t ws_size,
                              hipStream_t stream)
{
  const float* Y  = (const float*)d_in[0];
  const float* W1 = (const float*)d_in[1];   // (D1, D0)
  const float* W2 = (const float*)d_in[2];   // (D2, D1)
  const float* W3 = (const float*)d_in[3];   // (D3, D2)
  const float* b1 = (const float*)d_in[4];
  const float* b2 = (const float*)d_in[5];
  const float* b3 = (const float*)d_in[6];
  float* out = (float*)d_out;

  const size_t FSZ = (size_t)BATCH * M_AA * DTOT;       // floats
  float* F  = (float*)d_ws;
  float* X  = F + FSZ;
  float* N1 = X + FSZ;                                  // BATCH x D1

  const dim3 blk(256);
  const dim3 gElem((unsigned)(((size_t)BATCH * DTOT + 255) / 256));

  // n1 = relu(Y @ W1^T), computed once (z-independent part of f)
  gemm_wmma<0, true, false><<<dim3(D1 / NBLK, BATCH / MBLK), blk, 0, stream>>>(
      Y, D0, W1, D0, N1, D1, nullptr, D0);

  init_hist<<<gElem, blk, 0, stream>>>(X, F, N1);

  // F[:,1,512:1024] = relu(n1 @ W2^T)   (the z-dependent part of f(f0))
  gemm_wmma<0, true, false><<<dim3(D1 / NBLK, BATCH / MBLK), blk, 0, stream>>>(
      N1, D1, W2, D1, F + DTOT + D1, M_AA * DTOT, nullptr, D1);

  for (int k = 2; k < THRES; ++k) {
    const int slot = k % M_AA;
    const int n = (k < M_AA) ? k : M_AA;
    anderson_step<<<BATCH / 8, blk, 0, stream>>>(F, X, N1, n, slot);
    // n2 = relu(z1 @ W2^T) -> F[slot][512:1024]
    gemm_wmma<0, true, false><<<dim3(D1 / NBLK, BATCH / MBLK), blk, 0, stream>>>(
        X + slot * DTOT, M_AA * DTOT, W2, D1, F + slot * DTOT + D1, M_AA * DTOT,
        nullptr, D1);
    // n3 = relu(z2 @ W3^T) -> F[slot][1024:1536]
    gemm_wmma<0, true, false><<<dim3(D2 / NBLK, BATCH / MBLK), blk, 0, stream>>>(
        X + slot * DTOT + D1, M_AA * DTOT, W3, D2, F + slot * DTOT + D1 + D2,
        M_AA * DTOT, nullptr, D2);
  }

  // z* = F[:, (THRES-1)%m = 4, :]
  const float* zf = F + 4 * DTOT;
  // eta1 = z1 @ W1 + b1  (NN: W1 is (512,1024) K-major-by-rows)
  gemm_wmma<1, false, true><<<dim3(D0 / NBLK, BATCH / MBLK), blk, 0, stream>>>(
      zf, M_AA * DTOT, W1, D0, out, OUTW, b1, D1);
  // eta2 = z2 @ W2 + b2
  gemm_wmma<1, false, true><<<dim3(D1 / NBLK, BATCH / MBLK), blk, 0, stream>>>(
      zf + D1, M_AA * DTOT, W2, D1, out + D0, OUTW, b2, D2);
  // eta3 = z3 @ W3 + b3
  gemm_wmma<1, false, true><<<dim3(D2 / NBLK, BATCH / MBLK), blk, 0, stream>>>(
      zf + D1 + D2, M_AA * DTOT, W3, D2, out + D0 + D1, OUTW, b3, D3);
  // out[:, 2048:3584] = z*
  copy_zstar<<<gElem, blk, 0, stream>>>(F, out);
}